// T5Attention_90606630077241
// MI455X (gfx1250) — compile-verified
//
#include <hip/hip_runtime.h>

// ---------------------------------------------------------------------------
// T5 encoder self-attention for MI455X (gfx1250, wave32).
//  * All matmuls on V_WMMA_F32_16X16X4_F32 (CDNA5 fp32 tensor path, matches
//    the fp32 reference numerics).
//  * All global->LDS tile staging via the Tensor Data Mover
//    (tensor_load_to_lds + s_wait_tensorcnt): no VGPR round trip, hardware
//    row padding (pad_interval/pad_amount) gives the bank-conflict-free
//    LDS strides directly.
//  * Flash-style attention: 256 MB score tensor never exists; online softmax.
// ---------------------------------------------------------------------------

typedef __attribute__((ext_vector_type(2))) float v2f;
typedef __attribute__((ext_vector_type(8))) float v8f;
typedef __attribute__((ext_vector_type(4))) unsigned int u32x4;
typedef __attribute__((ext_vector_type(8))) int i32x8;
typedef __attribute__((ext_vector_type(4))) int i32x4;

#define S_LEN   1024
#define D_MODEL 512
#define NHEAD   8
#define DKV     64
#define BATCH   8
#define TBL     (2 * S_LEN - 1)   // 2047 relative positions

#if defined(__gfx1250__) && __has_builtin(__builtin_amdgcn_tensor_load_to_lds)
#define USE_TDM 1
#else
#define USE_TDM 0
#endif

static __device__ __forceinline__ v8f wmma_f32(v2f a, v2f b, v8f c) {
  // D(16x16) = A(16x4) * B(4x16) + C, fp32 end to end.
  return __builtin_amdgcn_wmma_f32_16x16x4_f32(
      /*neg_a=*/false, a, /*neg_b=*/false, b,
      /*c_mod=*/(short)0, c, /*reuse_a=*/false, /*reuse_b=*/false);
}

#if USE_TDM
// ---------------------------------------------------------------------------
// TDM: DMA a 2D fp32 tile (tile_h rows x tile_w elems, global row stride
// row_stride elems) into LDS at byte offset lds_addr, inserting
// (pad_amount_code+1) DWORDs of padding after every 2<<pad_interval_code
// DWORDs stored (per CDNA5 D# spec, groups 0/1; groups 2/3 zero => 2D).
// tensor_dim == tile_dim so no OOB path is ever taken.
// Issue from ONE wave only; completion via s_wait_tensorcnt + barrier.
// This toolchain exposes the 6-arg builtin:
//   (uint32x4 g0, int32x8 g1, int32x4 g2, int32x4 g3, int32x8 extra, i32 cpol)
// ---------------------------------------------------------------------------
static __device__ __forceinline__ void tdm_load_2d_f32(
    unsigned lds_addr, const float* gptr,
    unsigned tile_w, unsigned tile_h, unsigned row_stride,
    unsigned pad_interval_code, unsigned pad_amount_code) {
  unsigned long long ga = (unsigned long long)(uintptr_t)gptr;
  u32x4 g0;
  g0[0] = 1u;                                   // count=1, is_restore=0
  g0[1] = lds_addr;                             // LDS byte address
  g0[2] = (unsigned)ga;                         // global_addr[31:0]
  g0[3] = (unsigned)((ga >> 32) & 0x1FFFFFFull) // global_addr[56:32]
        | (2u << 30);                           // type=2 ("image")
  i32x8 g1;
  g1[0] = (int)((2u << 16)                      // data_size = 4 bytes
              | (1u << 20)                      // pad_enable
              | (pad_interval_code << 22)
              | (pad_amount_code << 25));       // wg_mask=0 (no cluster)
  g1[1] = (int)(tile_w << 16);                  // tensor_dim0 = tile_w
  g1[2] = (int)(tile_h << 16);                  // tensor_dim1 = tile_h
  g1[3] = (int)(tile_w << 16);                  // tile_dim0  = tile_w
  g1[4] = (int)tile_h;                          // tile_dim1; tile_dim2 = 0
  g1[5] = (int)row_stride;                      // tensor_dim0_stride[31:0]
  g1[6] = 0;                                    // stride hi / dim1_stride lo
  g1[7] = 0;
  i32x4 gz4 = {0, 0, 0, 0};                     // groups 2/3: unused (2D)
  i32x8 gz8 = {0, 0, 0, 0, 0, 0, 0, 0};
  __builtin_amdgcn_tensor_load_to_lds(g0, g1, gz4, gz4, gz8, 0);
}
static __device__ __forceinline__ unsigned lds_off(const void* p) {
  return (unsigned)(uintptr_t)p;  // generic LDS ptr: low 32 bits = LDS offset
}
#endif

// ---------------------------------------------------------------------------
// Kernel 1: bias lookup table: table[h][d + S-1] for d = mem - ctx.
// ---------------------------------------------------------------------------
__global__ void bias_table_kernel(const float* __restrict__ rel_bias,
                                  float* __restrict__ table) {
  int i = blockIdx.x * blockDim.x + threadIdx.x;
  if (i >= TBL) return;
  int d = i - (S_LEN - 1);
  int bucket = (d > 0) ? 16 : 0;
  int rp = d < 0 ? -d : d;
  int bv;
  if (rp < 8) {
    bv = rp;
  } else {
    float l = logf((float)rp * 0.125f) * (8.0f / logf(16.0f));
    int large = 8 + (int)l;
    bv = large < 15 ? large : 15;
  }
  bucket += bv;
#pragma unroll
  for (int h = 0; h < NHEAD; ++h)
    table[h * TBL + i] = rel_bias[bucket * NHEAD + h];
}

// ---------------------------------------------------------------------------
// Kernel 2: fp32 GEMM C[MxN] = A[MxK]*B[KxN]. 128 threads (4 waves),
// 64x64 tile; wave w owns rows [16w,16w+16). TDM stages A (64x32, stride
// 36 = 32+4 pad) and B (32x64, stride 68 = 64+4 pad) tiles.
// ---------------------------------------------------------------------------
__global__ __launch_bounds__(128) void gemm_wmma_f32(
    const float* __restrict__ A, const float* __restrict__ B,
    float* __restrict__ C, int M, int N, int K, int head_split) {
  __shared__ float As[64 * 36];
  __shared__ float Bs[32 * 68];

  const int tid  = threadIdx.x;
  const int w    = tid >> 5;
  const int lane = tid & 31;
  const int ml   = lane & 15;
  const int hf   = lane >> 4;
  const int m0   = blockIdx.y * 64;
  const int n0   = blockIdx.x * 64;

  const v8f vz = {0.f, 0.f, 0.f, 0.f, 0.f, 0.f, 0.f, 0.f};
  v8f acc[4];
#pragma unroll
  for (int f = 0; f < 4; ++f) acc[f] = vz;

  for (int kb = 0; kb < K; kb += 32) {
    __syncthreads();
#if USE_TDM
    if (w == 0) {
      // A: 64 rows x 32 elems, pad 32->36 (interval code 4 = 32 DW, amt 3 = 4 DW)
      tdm_load_2d_f32(lds_off(As), A + (size_t)m0 * K + kb, 32, 64, (unsigned)K, 4, 3);
      // B: 32 rows x 64 elems, pad 64->68 (interval code 5 = 64 DW, amt 3 = 4 DW)
      tdm_load_2d_f32(lds_off(Bs), B + (size_t)kb * N + n0, 64, 32, (unsigned)N, 5, 3);
      __builtin_amdgcn_s_wait_tensorcnt(0);
    }
#else
    for (int i = tid; i < 512; i += 128) {
      int r = i >> 3, c4 = i & 7;
      *(float4*)(As + r * 36 + c4 * 4) =
          *(const float4*)(A + (size_t)(m0 + r) * K + kb + c4 * 4);
    }
    for (int i = tid; i < 512; i += 128) {
      int r = i >> 4, c4 = i & 15;
      *(float4*)(Bs + r * 68 + c4 * 4) =
          *(const float4*)(B + (size_t)(kb + r) * N + n0 + c4 * 4);
    }
#endif
    __syncthreads();
#pragma unroll
    for (int t = 0; t < 8; ++t) {                   // kk = 4*t
      v2f a;
      a.x = As[(16 * w + ml) * 36 + 4 * t + 2 * hf];
      a.y = As[(16 * w + ml) * 36 + 4 * t + 2 * hf + 1];
#pragma unroll
      for (int f = 0; f < 4; ++f) {
        v2f bf;
        bf.x = Bs[(4 * t + 2 * hf) * 68 + f * 16 + ml];
        bf.y = Bs[(4 * t + 2 * hf + 1) * 68 + f * 16 + ml];
        acc[f] = wmma_f32(a, bf, acc[f]);
      }
    }
  }

#pragma unroll
  for (int f = 0; f < 4; ++f) {
#pragma unroll
    for (int r = 0; r < 8; ++r) {
      int gm = m0 + 16 * w + r + 8 * hf;
      int gn = n0 + f * 16 + ml;
      float val = acc[f][r];
      if (head_split) {  // [B*S, D] -> [B, H, S, 64]
        int bb = gm >> 10, ss = gm & 1023, hh = gn >> 6, cc = gn & 63;
        C[(((size_t)bb * NHEAD + hh) * S_LEN + ss) * DKV + cc] = val;
      } else {
        C[(size_t)gm * N + gn] = val;
      }
    }
  }
}

// ---------------------------------------------------------------------------
// Kernel 3: flash attention per (b, h, 64-row q tile). TDM stages Q once and
// K/V tiles per 64-key block; online softmax with 16-lane-half shuffle
// reductions matched to the WMMA C-fragment layout; P re-laid out through
// per-wave LDS (C-frag -> A-frag) for the P@V WMMA.
// ---------------------------------------------------------------------------
__global__ __launch_bounds__(128) void flash_attn_wmma(
    const float* __restrict__ Q, const float* __restrict__ Km,
    const float* __restrict__ V, const float* __restrict__ table,
    float* __restrict__ ctx) {
  __shared__ float Ks[64 * 68];
  __shared__ float Vs[64 * 68];
  __shared__ float QPs[64 * 68];  // Q tile in prologue; reused as P tiles

  const int tid  = threadIdx.x;
  const int w    = tid >> 5;
  const int lane = tid & 31;
  const int ml   = lane & 15;
  const int hf   = lane >> 4;
  const int bh   = blockIdx.y;
  const int b    = bh >> 3;
  const int h    = bh & 7;
  const int q0   = blockIdx.x * 64;

  const float* Qp = Q + (size_t)bh * S_LEN * DKV;
  const float* Kp = Km + (size_t)bh * S_LEN * DKV;
  const float* Vp = V + (size_t)bh * S_LEN * DKV;
  const float* bias = table + h * TBL + (S_LEN - 1) - q0;

  // ---- stage Q tile, build per-wave A-fragments in registers -------------
#if USE_TDM
  if (w == 0) {
    tdm_load_2d_f32(lds_off(QPs), Qp + (size_t)q0 * DKV, 64, 64, 64, 5, 3);
    __builtin_amdgcn_s_wait_tensorcnt(0);
  }
#else
  for (int i = tid; i < 1024; i += 128) {
    int r = i >> 4, c4 = i & 15;
    *(float4*)(QPs + r * 68 + c4 * 4) =
        *(const float4*)(Qp + (size_t)(q0 + r) * DKV + c4 * 4);
  }
#endif
  __syncthreads();
  v2f qa[16];
#pragma unroll
  for (int t = 0; t < 16; ++t) {
    qa[t].x = QPs[(16 * w + ml) * 68 + 4 * t + 2 * hf];
    qa[t].y = QPs[(16 * w + ml) * 68 + 4 * t + 2 * hf + 1];
  }
  __syncthreads();  // Q region now free for reuse as P

  const v8f vz = {0.f, 0.f, 0.f, 0.f, 0.f, 0.f, 0.f, 0.f};
  v8f cacc[4];
#pragma unroll
  for (int f = 0; f < 4; ++f) cacc[f] = vz;
  float rmax[8], rsum[8];
#pragma unroll
  for (int r = 0; r < 8; ++r) { rmax[r] = -__builtin_inff(); rsum[r] = 0.f; }

  float* Pw = QPs + w * 16 * 68;  // per-wave 16x64 P tile

  for (int kb = 0; kb < S_LEN; kb += 64) {
    __syncthreads();
#if USE_TDM
    if (w == 0) {
      tdm_load_2d_f32(lds_off(Ks), Kp + (size_t)kb * DKV, 64, 64, 64, 5, 3);
      tdm_load_2d_f32(lds_off(Vs), Vp + (size_t)kb * DKV, 64, 64, 64, 5, 3);
      __builtin_amdgcn_s_wait_tensorcnt(0);
    }
#else
    for (int i = tid; i < 1024; i += 128) {
      int r = i >> 4, c4 = i & 15;
      *(float4*)(Ks + r * 68 + c4 * 4) =
          *(const float4*)(Kp + (size_t)(kb + r) * DKV + c4 * 4);
      *(float4*)(Vs + r * 68 + c4 * 4) =
          *(const float4*)(Vp + (size_t)(kb + r) * DKV + c4 * 4);
    }
#endif
    __syncthreads();

    // ---- S = q * k^T (B-frag reads k[n][kk] from LDS: implicit transpose)
    v8f sc[4];
#pragma unroll
    for (int f = 0; f < 4; ++f) sc[f] = vz;
#pragma unroll
    for (int t = 0; t < 16; ++t) {  // kk = 4*t over dk=64
      v2f a = qa[t];
#pragma unroll
      for (int f = 0; f < 4; ++f) {
        v2f bf;
        bf.x = Ks[(f * 16 + ml) * 68 + 4 * t + 2 * hf];
        bf.y = Ks[(f * 16 + ml) * 68 + 4 * t + 2 * hf + 1];
        sc[f] = wmma_f32(a, bf, sc[f]);
      }
    }

    // ---- add relative-position bias; block row max ----------------------
    float bmax[8];
#pragma unroll
    for (int r = 0; r < 8; ++r) bmax[r] = -__builtin_inff();
#pragma unroll
    for (int f = 0; f < 4; ++f) {
      int n = kb + f * 16 + ml;
#pragma unroll
      for (int r = 0; r < 8; ++r) {
        int mloc = 16 * w + r + 8 * hf;
        float s = sc[f][r] + bias[n - mloc];
        sc[f][r] = s;
        bmax[r] = fmaxf(bmax[r], s);
      }
    }
#pragma unroll
    for (int r = 0; r < 8; ++r) {
#pragma unroll
      for (int off = 1; off < 16; off <<= 1)  // reduce within 16-lane half
        bmax[r] = fmaxf(bmax[r], __shfl_xor(bmax[r], off, 32));
    }

    // ---- online softmax update ------------------------------------------
    float lsum[8];
#pragma unroll
    for (int r = 0; r < 8; ++r) {
      float mnew = fmaxf(rmax[r], bmax[r]);
      float corr = __expf(rmax[r] - mnew);
      rmax[r] = mnew;
      rsum[r] *= corr;
#pragma unroll
      for (int f = 0; f < 4; ++f) cacc[f][r] *= corr;
      lsum[r] = 0.f;
    }
#pragma unroll
    for (int f = 0; f < 4; ++f) {
#pragma unroll
      for (int r = 0; r < 8; ++r) {
        float p = __expf(sc[f][r] - rmax[r]);
        lsum[r] += p;
        Pw[(r + 8 * hf) * 68 + f * 16 + ml] = p;  // C-layout -> LDS
      }
    }
#pragma unroll
    for (int r = 0; r < 8; ++r) {
#pragma unroll
      for (int off = 1; off < 16; off <<= 1)
        lsum[r] += __shfl_xor(lsum[r], off, 32);
      rsum[r] += lsum[r];
    }
    __syncthreads();

    // ---- ctx += P(16x64) * V(64x64) -------------------------------------
#pragma unroll
    for (int t = 0; t < 16; ++t) {  // kk = 4*t over 64 keys
      v2f a;
      a.x = Pw[ml * 68 + 4 * t + 2 * hf];       // A-layout read from LDS
      a.y = Pw[ml * 68 + 4 * t + 2 * hf + 1];
#pragma unroll
      for (int f = 0; f < 4; ++f) {
        v2f bf;
        bf.x = Vs[(4 * t + 2 * hf) * 68 + f * 16 + ml];
        bf.y = Vs[(4 * t + 2 * hf + 1) * 68 + f * 16 + ml];
        cacc[f] = wmma_f32(a, bf, cacc[f]);
      }
    }
  }

  // ---- normalize, write ctx in [B, S, H*64] layout -----------------------
#pragma unroll
  for (int f = 0; f < 4; ++f) {
#pragma unroll
    for (int r = 0; r < 8; ++r) {
      int mloc = 16 * w + r + 8 * hf;
      int srow = q0 + mloc;
      int col  = h * DKV + f * 16 + ml;
      ctx[(size_t)(b * S_LEN + srow) * D_MODEL + col] = cacc[f][r] / rsum[r];
    }
  }
}

// ---------------------------------------------------------------------------
// Launcher. Workspace (floats): q,k,v,ctx[4*4194304] + table[16376] ~64.1 MB.
// ---------------------------------------------------------------------------
extern "C" void kernel_launch(void* const* d_in, const int* in_sizes, int n_in,
                              void* d_out, int out_size, void* d_ws, size_t ws_size,
                              hipStream_t stream) {
  (void)in_sizes; (void)n_in; (void)out_size; (void)ws_size;
  const float* x  = (const float*)d_in[0];
  const float* Wq = (const float*)d_in[1];
  const float* Wk = (const float*)d_in[2];
  const float* Wv = (const float*)d_in[3];
  const float* Wo = (const float*)d_in[4];
  const float* rb = (const float*)d_in[5];
  float* out = (float*)d_out;

  const size_t QKV = (size_t)BATCH * NHEAD * S_LEN * DKV;
  const size_t CTX = (size_t)BATCH * S_LEN * D_MODEL;
  float* ws  = (float*)d_ws;
  float* q   = ws;
  float* k   = q + QKV;
  float* v   = k + QKV;
  float* cx  = v + QKV;
  float* tbl = cx + CTX;

  bias_table_kernel<<<dim3((TBL + 255) / 256), dim3(256), 0, stream>>>(rb, tbl);

  dim3 gg(D_MODEL / 64, (BATCH * S_LEN) / 64);  // (8, 128)
  gemm_wmma_f32<<<gg, 128, 0, stream>>>(x, Wq, q, BATCH * S_LEN, D_MODEL, D_MODEL, 1);
  gemm_wmma_f32<<<gg, 128, 0, stream>>>(x, Wk, k, BATCH * S_LEN, D_MODEL, D_MODEL, 1);
  gemm_wmma_f32<<<gg, 128, 0, stream>>>(x, Wv, v, BATCH * S_LEN, D_MODEL, D_MODEL, 1);

  flash_attn_wmma<<<dim3(S_LEN / 64, BATCH * NHEAD), 128, 0, stream>>>(q, k, v, tbl, cx);

  gemm_wmma_f32<<<gg, 128, 0, stream>>>(cx, Wo, out, BATCH * S_LEN, D_MODEL, D_MODEL, 0);
}